// get_loss_90864328114248
// MI455X (gfx1250) — compile-verified
//
#include <hip/hip_runtime.h>
#include <hip/hip_bf16.h>

#define BB 8
#define NN 4096
#define MM 4096
#define CC 16
#define BIGF 1.0e10f

typedef __attribute__((ext_vector_type(2))) float v2f;
typedef __attribute__((ext_vector_type(8))) float v8f;

// ---------------------------------------------------------------------------
// Pass 0: pred class (argmax of int-truncated log-probs, first-max-wins, as in
// the reference's argmax(pred_label.astype(int32))) and seg gather.
// ---------------------------------------------------------------------------
__global__ __launch_bounds__(256) void prep_kernel(
    const float* __restrict__ pred_label, const int* __restrict__ true_label,
    int* __restrict__ pred_cls, float* __restrict__ seg) {
  int i = blockIdx.x * 256 + threadIdx.x;  // [0, B*N)
  const float* lp = pred_label + (size_t)i * CC;
  int best_j = 0;
  int best_v = (int)lp[0];
  #pragma unroll
  for (int j = 1; j < CC; ++j) {
    int v = (int)lp[j];
    if (v > best_v) { best_v = v; best_j = j; }
  }
  pred_cls[i] = best_j;
  int tl = true_label[i];
  seg[i] = lp[tl];
}

// ---------------------------------------------------------------------------
// Pass A: rows = pred points, cols = gt points.
// Each wave owns a 16-row pred tile, loops over all gt points in 16-col tiles,
// computes dot products with V_WMMA_F32_16X16X4_F32 (K=3 padded to 4), forms
// D = p2 + g2 - 2*dot, and keeps per-row per-gt-class minima in LDS.
// Outputs cmin1[b,n,c] and d1[b,n].
// ---------------------------------------------------------------------------
__global__ __launch_bounds__(256) void pass_pred_rows(
    const float* __restrict__ pred_xyz, const float* __restrict__ gt_xyzl,
    float* __restrict__ cmin1, float* __restrict__ d1) {
  __shared__ float Dtile[8][256];   // per-wave 16x16 dot tile
  __shared__ float g2col[8][16];
  __shared__ int   clscol[8][16];
  __shared__ float cmin[8][256];    // per-wave [row][class] minima

  const int tid  = threadIdx.x;
  const int wave = tid >> 5;
  const int lane = tid & 31;
  const int half = lane >> 4;
  const int lrow = lane & 15;
  const int tilesPerB = NN / 128;
  const int b    = blockIdx.x / tilesPerB;
  const int tile = blockIdx.x % tilesPerB;
  const int n    = tile * 128 + wave * 16 + lrow;

  // A-matrix: 16x4 fp32, lanes 0-15 -> (K0,K1)=(x,y); lanes 16-31 -> (K2,K3)=(z,0)
  const float px = pred_xyz[((size_t)(b * NN + n)) * 3 + 0];
  const float py = pred_xyz[((size_t)(b * NN + n)) * 3 + 1];
  const float pz = pred_xyz[((size_t)(b * NN + n)) * 3 + 2];
  const float p2 = px * px + py * py + pz * pz;
  v2f a;
  a.x = half ? pz : px;
  a.y = half ? 0.0f : py;

  for (int i = lane; i < 256; i += 32) cmin[wave][i] = BIGF;

  for (int m0 = 0; m0 < MM; m0 += 16) {
    const float4 g = ((const float4*)gt_xyzl)[(size_t)b * MM + m0 + lrow];
    v2f bv;
    bv.x = half ? g.z : g.x;
    bv.y = half ? 0.0f : g.y;
    if (lane < 16) {
      g2col[wave][lrow]  = g.x * g.x + g.y * g.y + g.z * g.z;
      clscol[wave][lrow] = (int)g.w;
    }
    v8f acc = {};
    acc = __builtin_amdgcn_wmma_f32_16x16x4_f32(
        false, a, false, bv, (short)0, acc, false, false);
    #pragma unroll
    for (int r = 0; r < 8; ++r)
      Dtile[wave][(r + 8 * half) * 16 + lrow] = acc[r];
    __syncthreads();
    if (lane < 16) {
      #pragma unroll
      for (int c = 0; c < 16; ++c) {
        float d = p2 + g2col[wave][c] - 2.0f * Dtile[wave][lrow * 16 + c];
        int j = clscol[wave][c];
        float cur = cmin[wave][lrow * 16 + j];
        if (d < cur) cmin[wave][lrow * 16 + j] = d;
      }
    }
    __syncthreads();
  }

  if (lane < 16) {
    float dmin = BIGF;
    #pragma unroll
    for (int j = 0; j < CC; ++j) {
      float v = cmin[wave][lrow * 16 + j];
      cmin1[((size_t)(b * NN + n)) * CC + j] = v;
      dmin = fminf(dmin, v);
    }
    d1[b * NN + n] = dmin;
  }
}

// ---------------------------------------------------------------------------
// Pass B: rows = gt points, cols = pred points (symmetric).
// Outputs cmin2[b,m,c] and d2[b,m].
// ---------------------------------------------------------------------------
__global__ __launch_bounds__(256) void pass_gt_rows(
    const float* __restrict__ gt_xyzl, const float* __restrict__ pred_xyz,
    const int* __restrict__ pred_cls,
    float* __restrict__ cmin2, float* __restrict__ d2) {
  __shared__ float Dtile[8][256];
  __shared__ float p2col[8][16];
  __shared__ int   clscol[8][16];
  __shared__ float cmin[8][256];

  const int tid  = threadIdx.x;
  const int wave = tid >> 5;
  const int lane = tid & 31;
  const int half = lane >> 4;
  const int lrow = lane & 15;
  const int tilesPerB = MM / 128;
  const int b    = blockIdx.x / tilesPerB;
  const int tile = blockIdx.x % tilesPerB;
  const int m    = tile * 128 + wave * 16 + lrow;

  const float4 g = ((const float4*)gt_xyzl)[(size_t)b * MM + m];
  const float g2 = g.x * g.x + g.y * g.y + g.z * g.z;
  v2f a;
  a.x = half ? g.z : g.x;
  a.y = half ? 0.0f : g.y;

  for (int i = lane; i < 256; i += 32) cmin[wave][i] = BIGF;

  for (int n0 = 0; n0 < NN; n0 += 16) {
    const int nc = n0 + lrow;
    const float px = pred_xyz[((size_t)(b * NN + nc)) * 3 + 0];
    const float py = pred_xyz[((size_t)(b * NN + nc)) * 3 + 1];
    const float pz = pred_xyz[((size_t)(b * NN + nc)) * 3 + 2];
    v2f bv;
    bv.x = half ? pz : px;
    bv.y = half ? 0.0f : py;
    if (lane < 16) {
      p2col[wave][lrow]  = px * px + py * py + pz * pz;
      clscol[wave][lrow] = pred_cls[b * NN + nc];
    }
    v8f acc = {};
    acc = __builtin_amdgcn_wmma_f32_16x16x4_f32(
        false, a, false, bv, (short)0, acc, false, false);
    #pragma unroll
    for (int r = 0; r < 8; ++r)
      Dtile[wave][(r + 8 * half) * 16 + lrow] = acc[r];
    __syncthreads();
    if (lane < 16) {
      #pragma unroll
      for (int c = 0; c < 16; ++c) {
        float d = g2 + p2col[wave][c] - 2.0f * Dtile[wave][lrow * 16 + c];
        int j = clscol[wave][c];
        float cur = cmin[wave][lrow * 16 + j];
        if (d < cur) cmin[wave][lrow * 16 + j] = d;
      }
    }
    __syncthreads();
  }

  if (lane < 16) {
    float dmin = BIGF;
    #pragma unroll
    for (int j = 0; j < CC; ++j) {
      float v = cmin[wave][lrow * 16 + j];
      cmin2[((size_t)(b * MM + m)) * CC + j] = v;
      dmin = fminf(dmin, v);
    }
    d2[b * MM + m] = dmin;
  }
}

// ---------------------------------------------------------------------------
// Pass C1: per (b, class) masked sums/counts. One block per (b,j).
// Deterministic strided accumulation + tree reduction.
// ---------------------------------------------------------------------------
__global__ __launch_bounds__(256) void class_sums(
    const float* __restrict__ cmin1, const float* __restrict__ cmin2,
    const int* __restrict__ pred_cls, const float* __restrict__ gt_xyzl,
    float* __restrict__ sum1, float* __restrict__ cnt1,
    float* __restrict__ sum2, float* __restrict__ cnt2) {
  __shared__ float red[256];
  const int b = blockIdx.x >> 4;
  const int j = blockIdx.x & 15;
  const int tid = threadIdx.x;

  float s1 = 0.f, c1 = 0.f, s2 = 0.f, c2 = 0.f;
  for (int n = tid; n < NN; n += 256) {
    if (pred_cls[b * NN + n] == j) {
      s1 += cmin1[((size_t)(b * NN + n)) * CC + j];
      c1 += 1.0f;
    }
  }
  for (int m = tid; m < MM; m += 256) {
    int gj = (int)gt_xyzl[((size_t)(b * MM + m)) * 4 + 3];
    if (gj == j) {
      s2 += cmin2[((size_t)(b * MM + m)) * CC + j];
      c2 += 1.0f;
    }
  }
  float vals[4] = {s1, c1, s2, c2};
  float out[4];
  #pragma unroll
  for (int k = 0; k < 4; ++k) {
    red[tid] = vals[k];
    __syncthreads();
    for (int s = 128; s > 0; s >>= 1) {
      if (tid < s) red[tid] += red[tid + s];
      __syncthreads();
    }
    out[k] = red[0];
    __syncthreads();
  }
  if (tid == 0) {
    sum1[blockIdx.x] = out[0];
    cnt1[blockIdx.x] = out[1];
    sum2[blockIdx.x] = out[2];
    cnt2[blockIdx.x] = out[3];
  }
}

// ---------------------------------------------------------------------------
// Pass C2: final scalars.
// ---------------------------------------------------------------------------
__global__ __launch_bounds__(256) void finalize_kernel(
    const float* __restrict__ d1, const float* __restrict__ d2,
    const float* __restrict__ seg,
    const float* __restrict__ sum1, const float* __restrict__ cnt1,
    const float* __restrict__ sum2, const float* __restrict__ cnt2,
    const int* __restrict__ epoch_p, float* __restrict__ out) {
  __shared__ float red[256];
  const int tid = threadIdx.x;

  float sd1 = 0.f, sd2 = 0.f, ss = 0.f;
  for (int i = tid; i < BB * NN; i += 256) sd1 += d1[i];
  for (int i = tid; i < BB * MM; i += 256) sd2 += d2[i];
  for (int i = tid; i < BB * NN; i += 256) ss += seg[i];

  float vals[3] = {sd1, sd2, ss};
  float tot[3];
  #pragma unroll
  for (int k = 0; k < 3; ++k) {
    red[tid] = vals[k];
    __syncthreads();
    for (int s = 128; s > 0; s >>= 1) {
      if (tid < s) red[tid] += red[tid + s];
      __syncthreads();
    }
    tot[k] = red[0];
    __syncthreads();
  }

  if (tid == 0) {
    const float loss_cham = (tot[0] + tot[1]) / (float)BB;
    const float loss_seg  = -tot[2] / (float)(BB * NN);
    const float w[CC] = {1.f, 1.f, 1.f, 1.f, 1.f, 0.01f, 1.f, 1.f,
                         1.f, 0.01f, 1.f, 1.f, 1.f, 1.f, 1.f, 1.f};
    float loss_sem = 0.0f;
    for (int j = 0; j < CC; ++j) {
      float lc = 0.f, tc = 0.f;
      for (int b = 0; b < BB; ++b) {
        int idx = b * CC + j;
        float c1 = cnt1[idx], c2 = cnt2[idx];
        if (c1 > 0.0f && c2 > 0.0f) {
          lc += sum1[idx] / fmaxf(c1, 1.0f) + sum2[idx] / fmaxf(c2, 1.0f);
          tc += 1.0f;
        }
      }
      lc *= w[j];
      if (lc != 0.0f) loss_sem += lc / fmaxf(tc, 1.0f);
    }
    const int epoch = epoch_p[0];
    float loss, lsem_out;
    if (epoch >= 255) {
      lsem_out = loss_sem;
      loss = 0.02f * loss_cham + loss_sem + loss_seg;
    } else {
      lsem_out = 1000.0f;
      loss = 0.005f * loss_cham + loss_seg;
    }
    out[0] = loss;
    out[1] = loss_cham;
    out[2] = lsem_out;
    out[3] = loss_seg;
  }
}

// ---------------------------------------------------------------------------
extern "C" void kernel_launch(void* const* d_in, const int* in_sizes, int n_in,
                              void* d_out, int out_size, void* d_ws, size_t ws_size,
                              hipStream_t stream) {
  const float* pred_xyz   = (const float*)d_in[0];  // (B,N,3)
  const float* pred_label = (const float*)d_in[1];  // (B,N,C)
  const float* gt_xyzl    = (const float*)d_in[2];  // (B,M,4)
  const int*   true_label = (const int*)d_in[3];    // (B,N)
  const int*   epoch      = (const int*)d_in[4];    // scalar
  float* out = (float*)d_out;                       // 4 scalars

  char* ws = (char*)d_ws;
  float* cmin1 = (float*)(ws);                              // B*N*C floats (2 MB)
  float* cmin2 = (float*)(ws + 2097152);                    // B*M*C floats (2 MB)
  float* d1    = (float*)(ws + 4194304);                    // B*N floats
  float* d2    = (float*)(ws + 4194304 + 131072);           // B*M floats
  float* seg   = (float*)(ws + 4194304 + 262144);           // B*N floats
  int*   pcls  = (int*)  (ws + 4194304 + 393216);           // B*N ints
  float* sum1  = (float*)(ws + 4194304 + 524288);           // 128 floats
  float* cnt1  = sum1 + 128;
  float* sum2  = sum1 + 256;
  float* cnt2  = sum1 + 384;

  prep_kernel<<<(BB * NN) / 256, 256, 0, stream>>>(pred_label, true_label, pcls, seg);
  pass_pred_rows<<<BB * (NN / 128), 256, 0, stream>>>(pred_xyz, gt_xyzl, cmin1, d1);
  pass_gt_rows<<<BB * (MM / 128), 256, 0, stream>>>(gt_xyzl, pred_xyz, pcls, cmin2, d2);
  class_sums<<<BB * CC, 256, 0, stream>>>(cmin1, cmin2, pcls, gt_xyzl,
                                          sum1, cnt1, sum2, cnt2);
  finalize_kernel<<<1, 256, 0, stream>>>(d1, d2, seg, sum1, cnt1, sum2, cnt2,
                                         epoch, out);
}